// RGAT_27547920236878
// MI455X (gfx1250) — compile-verified
//
#include <hip/hip_runtime.h>
#include <math.h>

typedef __attribute__((ext_vector_type(2))) float v2f;
typedef __attribute__((ext_vector_type(8))) float v8f;

#define C_DIM 64
#define NEG_SLOPE 0.2f
#define RES_LAMBDA 0.5f
#define R_PAD 32

// Signed-float atomic max via int/uint ordering trick (valid with -inf init).
__device__ __forceinline__ void atomicMaxF(float* addr, float val) {
    if (val >= 0.0f) atomicMax((int*)addr, __float_as_int(val));
    else             atomicMin((unsigned int*)addr, __float_as_uint(val));
}

// Zero-padded copy of relation_emb into [32][64] so the WMMA loop needs no
// lane guards (no EXEC-mask branches around the matrix loads).
__global__ void pad_rel_kernel(const float* __restrict__ rel, float* __restrict__ relPad, int R) {
    int idx = blockIdx.x * blockDim.x + threadIdx.x;
    if (idx >= R_PAD * C_DIM) return;
    int r = idx / C_DIM;
    relPad[idx] = (r < R) ? rel[idx] : 0.0f;
}

// ---------------------------------------------------------------------------
// Wr[r][k] = sum_c W[k][c] * rel[r][c]   (r padded to 32 rows, k < 128)
// V_WMMA_F32_16X16X4_F32: M = relation tile (2), N = k tile (8), K = 64 in
// steps of 4. One wave per output tile, 16 waves in one block.
// A (16x4): lanes<16 hold K=0,1; lanes>=16 hold K=2,3 (ISA VGPR layout).
// B (4x16): same split with N = lane&15.  C/D: VGPR j -> M=j (+8 hi half).
// ---------------------------------------------------------------------------
__global__ void wr_wmma_kernel(const float* __restrict__ W, const float* __restrict__ relPad,
                               float* __restrict__ Wr) {
    int lane  = threadIdx.x & 31;
    int wave  = threadIdx.x >> 5;   // 0..15
    int rtile = wave >> 3;          // 0..1  (relation rows)
    int ktile = wave & 7;           // 0..7  (k columns)
    int half  = lane >> 4;          // 0/1
    int lm    = lane & 15;

    const v2f* relv = (const v2f*)relPad;
    const v2f* wv   = (const v2f*)W;
    int rrow = rtile * 16 + lm;     // relation index (M)
    int wrow = ktile * 16 + lm;     // W row = k index (N)

    v8f acc = {};
    for (int c0 = 0; c0 < C_DIM; c0 += 4) {
        int kb = c0 + half * 2;     // even
        v2f a = relv[(rrow * C_DIM + kb) >> 1];   // unconditional b64 load
        v2f b = wv[(wrow * C_DIM + kb) >> 1];     // unconditional b64 load
        acc = __builtin_amdgcn_wmma_f32_16x16x4_f32(
                  false, a, false, b, (short)0, acc, false, false);
    }
    #pragma unroll
    for (int j = 0; j < 8; ++j) {
        int m = rtile * 16 + j + half * 8;   // relation row (rows >= R junk, never read)
        int n = ktile * 16 + lm;             // k
        Wr[m * 128 + n] = acc[j];
    }
}

// agg = entSrc (residual folded into scatter init); m = -inf; s = 0
__global__ void init_hop_kernel(const float* __restrict__ entSrc, float* __restrict__ agg,
                                float* __restrict__ m, float* __restrict__ s, int N) {
    long long idx = (long long)blockIdx.x * blockDim.x + threadIdx.x;
    long long total = (long long)N * C_DIM;
    if (idx >= total) return;
    agg[idx] = entSrc[idx];
    if ((idx & (C_DIM - 1)) == 0) {
        long long i = idx >> 6;
        m[i] = __int_as_float(0xFF800000);   // -inf
        s[i] = 0.0f;
    }
}

// One wave per edge: e = dot(ent[head], Wr[r][0:64]) + dot(ent[tail], Wr[r][64:128])
__global__ void edge_logit_kernel(const float* __restrict__ ent, const float* __restrict__ Wr,
                                  const int* __restrict__ head, const int* __restrict__ tail,
                                  const int* __restrict__ etype,
                                  float* __restrict__ e_out, float* __restrict__ m, long long E) {
    int lane = threadIdx.x & 31;
    long long edge = (long long)blockIdx.x * (blockDim.x >> 5) + (threadIdx.x >> 5);
    if (edge >= E) return;
    long long h = head[edge];
    long long t = tail[edge];
    long long r = etype[edge];
    const float4* row = (lane < 16) ? (const float4*)(ent + h * C_DIM)
                                    : (const float4*)(ent + t * C_DIM);
    float4 v  = row[lane & 15];
    float4 wv = ((const float4*)(Wr + r * 128))[lane]; // lanes 0-15: k 0..63, 16-31: 64..127
    float p = v.x * wv.x + v.y * wv.y + v.z * wv.z + v.w * wv.w;
    #pragma unroll
    for (int off = 16; off > 0; off >>= 1) p += __shfl_xor(p, off, 32);
    if (lane == 0) {
        float ev = (p > 0.0f) ? p : NEG_SLOPE * p;   // LeakyReLU(0.2)
        e_out[edge] = ev;
        atomicMaxF(&m[h], ev);
    }
}

// ex = exp(e - m[head]); s[head] += ex  (e overwritten with ex)
__global__ void edge_exp_kernel(const int* __restrict__ head, float* __restrict__ e,
                                const float* __restrict__ m, float* __restrict__ s, long long E) {
    long long i = (long long)blockIdx.x * blockDim.x + threadIdx.x;
    if (i >= E) return;
    long long h = head[i];
    float ex = expf(e[i] - m[h]);
    e[i] = ex;
    atomicAdd(&s[h], ex);
}

// agg[head] += attn * ent[tail]; 64 consecutive threads = one edge (coalesced atomics)
__global__ void edge_scatter_kernel(const float* __restrict__ ent, const int* __restrict__ head,
                                    const int* __restrict__ tail, const float* __restrict__ ex,
                                    const float* __restrict__ s, float* __restrict__ agg, long long E) {
    long long idx = (long long)blockIdx.x * blockDim.x + threadIdx.x;
    long long edge = idx >> 6;
    int c = (int)(idx & (C_DIM - 1));
    if (edge >= E) return;
    long long h = head[edge], t = tail[edge];
    float attn = ex[edge] / s[h];
    atomicAdd(&agg[h * C_DIM + c], ent[t * C_DIM + c] * attn);
}

// One wave per node: entNew = agg / max(||agg||, 1e-12); out = 0.5*resSrc + entNew
__global__ void normalize_kernel(const float* __restrict__ agg, float* __restrict__ entNew,
                                 const float* __restrict__ resSrc, float* __restrict__ out, int N) {
    int lane = threadIdx.x & 31;
    long long i = (long long)blockIdx.x * (blockDim.x >> 5) + (threadIdx.x >> 5);
    if (i >= N) return;
    float2 v = ((const float2*)(agg + i * C_DIM))[lane];
    float ss = v.x * v.x + v.y * v.y;
    #pragma unroll
    for (int off = 16; off > 0; off >>= 1) ss += __shfl_xor(ss, off, 32);
    float inv = 1.0f / fmaxf(sqrtf(ss), 1e-12f);
    float2 en; en.x = v.x * inv; en.y = v.y * inv;
    ((float2*)(entNew + i * C_DIM))[lane] = en;
    float2 rv = ((const float2*)(resSrc + i * C_DIM))[lane];
    float2 ov; ov.x = RES_LAMBDA * rv.x + en.x; ov.y = RES_LAMBDA * rv.y + en.y;
    ((float2*)(out + i * C_DIM))[lane] = ov;
}

extern "C" void kernel_launch(void* const* d_in, const int* in_sizes, int n_in,
                              void* d_out, int out_size, void* d_ws, size_t ws_size,
                              hipStream_t stream) {
    const float* entity = (const float*)d_in[0];   // [N,64]
    const float* rel    = (const float*)d_in[1];   // [24,64]
    const float* W      = (const float*)d_in[2];   // [128,64]
    const int*   eidx   = (const int*)d_in[3];     // [2,E] (harness: integer -> int32)
    const int*   etype  = (const int*)d_in[4];     // [E]

    const int       N = in_sizes[0] / C_DIM;
    const int       R = in_sizes[1] / C_DIM;
    const long long E = (long long)in_sizes[4];
    const int* head = eidx;
    const int* tail = eidx + E;

    // workspace carve-up (256B aligned regions)
    char* ws = (char*)d_ws;
    size_t off = 0;
    #define WS_TAKE(TY, NAME, CNT) \
        TY* NAME = (TY*)(ws + off); off += (((size_t)(CNT) * sizeof(TY)) + 255) & ~(size_t)255;
    WS_TAKE(float, relPad, R_PAD * C_DIM)          // zero-padded relation_emb
    WS_TAKE(float, Wr,     R_PAD * 128)            // precomputed rel @ W^T (padded rows)
    WS_TAKE(float, entB,   (size_t)N * C_DIM)      // normalized embeddings after hop
    WS_TAKE(float, agg,    (size_t)N * C_DIM)
    WS_TAKE(float, ebuf,   (size_t)E)              // logits, then exp values
    WS_TAKE(float, mbuf,   (size_t)N)              // segment max
    WS_TAKE(float, sbuf,   (size_t)N)              // segment sum
    #undef WS_TAKE
    (void)ws_size; (void)n_in; (void)out_size;

    // One-time precompute shared by both hops
    pad_rel_kernel<<<(R_PAD * C_DIM + 255) / 256, 256, 0, stream>>>(rel, relPad, R);
    wr_wmma_kernel<<<1, 512, 0, stream>>>(W, relPad, Wr);

    const float* entCur = entity;      // embeddings at start of hop
    const float* resSrc = entity;      // res accumulator source
    float* outp = (float*)d_out;

    const long long totNC = (long long)N * C_DIM;
    for (int hop = 0; hop < 2; ++hop) {
        init_hop_kernel<<<(unsigned)((totNC + 255) / 256), 256, 0, stream>>>(
            entCur, agg, mbuf, sbuf, N);
        edge_logit_kernel<<<(unsigned)((E + 7) / 8), 256, 0, stream>>>(
            entCur, Wr, head, tail, etype, ebuf, mbuf, E);
        edge_exp_kernel<<<(unsigned)((E + 255) / 256), 256, 0, stream>>>(
            head, ebuf, mbuf, sbuf, E);
        edge_scatter_kernel<<<(unsigned)((E * C_DIM + 255) / 256), 256, 0, stream>>>(
            entCur, head, tail, ebuf, sbuf, agg, E);
        normalize_kernel<<<(unsigned)((N + 7) / 8), 256, 0, stream>>>(
            agg, entB, resSrc, outp, N);
        entCur = entB;
        resSrc = outp;
    }
}